// ADMM_Net_9620726743618
// MI455X (gfx1250) — compile-verified
//
#include <hip/hip_runtime.h>
#include <math.h>

typedef float v2f __attribute__((ext_vector_type(2)));
typedef float v8f __attribute__((ext_vector_type(8)));

#define WMMA4(A, B, C) \
  __builtin_amdgcn_wmma_f32_16x16x4_f32(false, (A), false, (B), (short)0, (C), false, false)

#define D0c 486
#define D1c 648
#define P0c 243
#define P1c 324
#define FH 972
#define FW 1296
#define NC 3
#define NITERS 10
#define HWSZ (FH * FW)          // 1259712
#define PLANE (NC * HWSZ)       // 3779136
#define CROPSZ (NC * D0c * D1c) // 944784
#define PIF 3.14159265358979323846f

// ---------------- table layout (floats, inside d_ws tables region) ---------
#define W36SZ (48 * 48)
#define W27SZ (32 * 32)
#define T36SZ (36 * 36)
#define T27SZ (27 * 36)
// forward then backward, real then imag
#define OFF_W36F_R 0
#define OFF_W36F_I (OFF_W36F_R + W36SZ)
#define OFF_W36B_R (OFF_W36F_I + W36SZ)
#define OFF_W36B_I (OFF_W36B_R + W36SZ)
#define OFF_W27F_R (OFF_W36B_I + W36SZ)
#define OFF_W27F_I (OFF_W27F_R + W27SZ)
#define OFF_W27B_R (OFF_W27F_I + W27SZ)
#define OFF_W27B_I (OFF_W27B_R + W27SZ)
#define OFF_T36F_R (OFF_W27B_I + W27SZ)
#define OFF_T36F_I (OFF_T36F_R + T36SZ)
#define OFF_T36B_R (OFF_T36F_I + T36SZ)
#define OFF_T36B_I (OFF_T36B_R + T36SZ)
#define OFF_T27F_R (OFF_T36B_I + T36SZ)
#define OFF_T27F_I (OFF_T27F_R + T27SZ)
#define OFF_T27B_R (OFF_T27F_I + T27SZ)
#define OFF_T27B_I (OFF_T27B_R + T27SZ)
#define TABLES_SZ (OFF_T27B_I + T27SZ)

// ---------------------------------------------------------------------------
// Table init: DFT matrices (zero-padded) and four-step twiddles.
// ---------------------------------------------------------------------------
__global__ void init_tables_k(float* tb) {
  int t = blockIdx.x * blockDim.x + threadIdx.x;
  if (t < W36SZ) {
    int r = t / 48, c = t % 48;
    float re = 0.f, im = 0.f;
    if (r < 36 && c < 36) {
      float ang = -2.f * PIF * (float)((r * c) % 36) / 36.f;
      re = cosf(ang);
      im = sinf(ang);
    }
    tb[OFF_W36F_R + t] = re;
    tb[OFF_W36F_I + t] = im;
    tb[OFF_W36B_R + t] = re;
    tb[OFF_W36B_I + t] = -im;
  }
  if (t < W27SZ) {
    int r = t / 32, c = t % 32;
    float re = 0.f, im = 0.f;
    if (r < 27 && c < 27) {
      float ang = -2.f * PIF * (float)((r * c) % 27) / 27.f;
      re = cosf(ang);
      im = sinf(ang);
    }
    tb[OFF_W27F_R + t] = re;
    tb[OFF_W27F_I + t] = im;
    tb[OFF_W27B_R + t] = re;
    tb[OFF_W27B_I + t] = -im;
  }
  if (t < T36SZ) {
    int n1 = t / 36, k2 = t % 36;
    float ang = -2.f * PIF * (float)((n1 * k2) % 1296) / 1296.f;
    float re = cosf(ang), im = sinf(ang);
    tb[OFF_T36F_R + t] = re;
    tb[OFF_T36F_I + t] = im;
    tb[OFF_T36B_R + t] = re;
    tb[OFF_T36B_I + t] = -im;
  }
  if (t < T27SZ) {
    int n1 = t / 36, k2 = t % 36;
    float ang = -2.f * PIF * (float)((n1 * k2) % 972) / 972.f;
    float re = cosf(ang), im = sinf(ang);
    tb[OFF_T27F_R + t] = re;
    tb[OFF_T27F_I + t] = im;
    tb[OFF_T27B_R + t] = re;
    tb[OFF_T27B_I + t] = -im;
  }
}

// ---------------------------------------------------------------------------
// 1296-point FFT along rows via four-step (36 x 36) with WMMA f32 16x16x4.
// One block per (channel,row). 288 threads = 9 waves; one wave per 16x16 tile
// of the padded 48x48 product. In-place capable (reads whole row to LDS
// before writing).
// ---------------------------------------------------------------------------
__global__ __launch_bounds__(288) void fft1296_k(
    const float* __restrict__ inR, const float* __restrict__ inI,
    float* __restrict__ outR, float* __restrict__ outI,
    const float* __restrict__ Wr, const float* __restrict__ Wi,
    const float* __restrict__ Tr, const float* __restrict__ Ti,
    int inReal, int outReal, float scale) {
  __shared__ float Ar[48][49], Ai[48][49];
  __shared__ float Br[48][49], Bi[48][49];
  __shared__ float LWr[48][49], LWi[48][49];

  const int tid = threadIdx.x;
  const int base = blockIdx.x * FW;

  for (int t = tid; t < 48 * 49; t += 288) {
    int r = t / 49, c = t % 49;
    Ar[r][c] = 0.f;
    Ai[r][c] = 0.f;
  }
  for (int t = tid; t < W36SZ; t += 288) {
    LWr[t / 48][t % 48] = Wr[t];
    LWi[t / 48][t % 48] = Wi[t];
  }
  __syncthreads();
  // gather: A[n1][n2] = x[n2*36 + n1]
  for (int t = tid; t < FW; t += 288) {
    int n1 = t % 36, n2 = t / 36;
    Ar[n1][n2] = inR[base + t];
    if (!inReal) Ai[n1][n2] = inI[base + t];
  }
  __syncthreads();

  const int wid = tid >> 5, lane = tid & 31;
  const int tm = wid / 3, tn = wid % 3;
  const int fr = lane & 15;
  const int kb0 = (lane >> 4) << 1;
  const int vmb = (lane >> 4) << 3;

  // ---- stage A: B = A * W36 (complex), K = 48 ----
  v8f cr = {0.f, 0.f, 0.f, 0.f, 0.f, 0.f, 0.f, 0.f};
  v8f ci = {0.f, 0.f, 0.f, 0.f, 0.f, 0.f, 0.f, 0.f};
#pragma unroll
  for (int k = 0; k < 48; k += 4) {
    int kb = k + kb0;
    int m = tm * 16 + fr;
    int n = tn * 16 + fr;
    v2f aR = {Ar[m][kb], Ar[m][kb + 1]};
    v2f aI = {Ai[m][kb], Ai[m][kb + 1]};
    v2f bR = {LWr[kb][n], LWr[kb + 1][n]};
    v2f bI = {LWi[kb][n], LWi[kb + 1][n]};
    v2f nbI = {-bI.x, -bI.y};
    cr = WMMA4(aR, bR, cr);
    cr = WMMA4(aI, nbI, cr);
    ci = WMMA4(aR, bI, ci);
    ci = WMMA4(aI, bR, ci);
  }
#pragma unroll
  for (int v = 0; v < 8; ++v) {
    int m = tm * 16 + v + vmb;
    int n = tn * 16 + fr;
    Br[m][n] = cr[v];
    Bi[m][n] = ci[v];
  }
  __syncthreads();

  // ---- twiddle: B[n1][k2] *= W_1296^{+-n1*k2} ----
  for (int t = tid; t < FW; t += 288) {
    int n1 = t / 36, k2 = t % 36;
    float tr = Tr[t], ti = Ti[t];
    float br = Br[n1][k2], bi = Bi[n1][k2];
    Br[n1][k2] = br * tr - bi * ti;
    Bi[n1][k2] = br * ti + bi * tr;
  }
  __syncthreads();

  // ---- stage C: C = W36 * B (complex), K = 48 ----
  cr = (v8f){0.f, 0.f, 0.f, 0.f, 0.f, 0.f, 0.f, 0.f};
  ci = (v8f){0.f, 0.f, 0.f, 0.f, 0.f, 0.f, 0.f, 0.f};
#pragma unroll
  for (int k = 0; k < 48; k += 4) {
    int kb = k + kb0;
    int m = tm * 16 + fr;
    int n = tn * 16 + fr;
    v2f aR = {LWr[m][kb], LWr[m][kb + 1]};
    v2f aI = {LWi[m][kb], LWi[m][kb + 1]};
    v2f bR = {Br[kb][n], Br[kb + 1][n]};
    v2f bI = {Bi[kb][n], Bi[kb + 1][n]};
    v2f nbI = {-bI.x, -bI.y};
    cr = WMMA4(aR, bR, cr);
    cr = WMMA4(aI, nbI, cr);
    ci = WMMA4(aR, bI, ci);
    ci = WMMA4(aI, bR, ci);
  }
#pragma unroll
  for (int v = 0; v < 8; ++v) {
    int k1 = tm * 16 + v + vmb;
    int k2 = tn * 16 + fr;
    if (k1 < 36 && k2 < 36) {
      int o = base + k1 * 36 + k2;
      outR[o] = scale * cr[v];
      if (!outReal) outI[o] = scale * ci[v];
    }
  }
}

// ---------------------------------------------------------------------------
// 972-point FFT along columns via four-step (27 x 36) with WMMA f32 16x16x4.
// One block per (channel,column). 192 threads = 6 waves (2x3 tiles of 32x48).
// ---------------------------------------------------------------------------
__global__ __launch_bounds__(192) void fft972_k(
    const float* __restrict__ inR, const float* __restrict__ inI,
    float* __restrict__ outR, float* __restrict__ outI,
    const float* __restrict__ W36r, const float* __restrict__ W36i,
    const float* __restrict__ W27r, const float* __restrict__ W27i,
    const float* __restrict__ Tr, const float* __restrict__ Ti,
    int inReal, int outReal, float scale) {
  __shared__ float Ar[32][49], Ai[32][49];
  __shared__ float Br[32][49], Bi[32][49];
  __shared__ float LW36r[48][49], LW36i[48][49];
  __shared__ float LW27r[32][33], LW27i[32][33];

  const int tid = threadIdx.x;
  const int id = blockIdx.x;
  const int ch = id / FW, w = id % FW;
  const int base = ch * HWSZ + w;

  for (int t = tid; t < 32 * 49; t += 192) {
    int r = t / 49, c = t % 49;
    Ar[r][c] = 0.f;
    Ai[r][c] = 0.f;
  }
  for (int t = tid; t < W36SZ; t += 192) {
    LW36r[t / 48][t % 48] = W36r[t];
    LW36i[t / 48][t % 48] = W36i[t];
  }
  for (int t = tid; t < W27SZ; t += 192) {
    LW27r[t / 32][t % 32] = W27r[t];
    LW27i[t / 32][t % 32] = W27i[t];
  }
  __syncthreads();
  // gather column: A[n1][n2] = x[n2*27 + n1]
  for (int t = tid; t < FH; t += 192) {
    int n1 = t % 27, n2 = t / 27;
    int g = base + t * FW;
    Ar[n1][n2] = inR[g];
    if (!inReal) Ai[n1][n2] = inI[g];
  }
  __syncthreads();

  const int wid = tid >> 5, lane = tid & 31;
  const int tm = wid / 3, tn = wid % 3; // tm in {0,1}, tn in {0,1,2}
  const int fr = lane & 15;
  const int kb0 = (lane >> 4) << 1;
  const int vmb = (lane >> 4) << 3;

  // ---- stage A: B(32x48) = A(32x48) * W36(48x48), K = 48 ----
  v8f cr = {0.f, 0.f, 0.f, 0.f, 0.f, 0.f, 0.f, 0.f};
  v8f ci = {0.f, 0.f, 0.f, 0.f, 0.f, 0.f, 0.f, 0.f};
#pragma unroll
  for (int k = 0; k < 48; k += 4) {
    int kb = k + kb0;
    int m = tm * 16 + fr;
    int n = tn * 16 + fr;
    v2f aR = {Ar[m][kb], Ar[m][kb + 1]};
    v2f aI = {Ai[m][kb], Ai[m][kb + 1]};
    v2f bR = {LW36r[kb][n], LW36r[kb + 1][n]};
    v2f bI = {LW36i[kb][n], LW36i[kb + 1][n]};
    v2f nbI = {-bI.x, -bI.y};
    cr = WMMA4(aR, bR, cr);
    cr = WMMA4(aI, nbI, cr);
    ci = WMMA4(aR, bI, ci);
    ci = WMMA4(aI, bR, ci);
  }
#pragma unroll
  for (int v = 0; v < 8; ++v) {
    int m = tm * 16 + v + vmb;
    int n = tn * 16 + fr;
    Br[m][n] = cr[v];
    Bi[m][n] = ci[v];
  }
  __syncthreads();

  // ---- twiddle: B[n1][k2] *= W_972^{+-n1*k2} (n1<27,k2<36) ----
  for (int t = tid; t < T27SZ; t += 192) {
    int n1 = t / 36, k2 = t % 36;
    float tr = Tr[t], ti = Ti[t];
    float br = Br[n1][k2], bi = Bi[n1][k2];
    Br[n1][k2] = br * tr - bi * ti;
    Bi[n1][k2] = br * ti + bi * tr;
  }
  __syncthreads();

  // ---- stage C: C(32x48) = W27(32x32) * B(32x48), K = 32 ----
  cr = (v8f){0.f, 0.f, 0.f, 0.f, 0.f, 0.f, 0.f, 0.f};
  ci = (v8f){0.f, 0.f, 0.f, 0.f, 0.f, 0.f, 0.f, 0.f};
#pragma unroll
  for (int k = 0; k < 32; k += 4) {
    int kb = k + kb0;
    int m = tm * 16 + fr;
    int n = tn * 16 + fr;
    v2f aR = {LW27r[m][kb], LW27r[m][kb + 1]};
    v2f aI = {LW27i[m][kb], LW27i[m][kb + 1]};
    v2f bR = {Br[kb][n], Br[kb + 1][n]};
    v2f bI = {Bi[kb][n], Bi[kb + 1][n]};
    v2f nbI = {-bI.x, -bI.y};
    cr = WMMA4(aR, bR, cr);
    cr = WMMA4(aI, nbI, cr);
    ci = WMMA4(aR, bI, ci);
    ci = WMMA4(aI, bR, ci);
  }
#pragma unroll
  for (int v = 0; v < 8; ++v) {
    int k1 = tm * 16 + v + vmb;
    int k2 = tn * 16 + fr;
    if (k1 < 27 && k2 < 36) {
      int g = base + (k1 * 36 + k2) * FW;
      outR[g] = scale * cr[v];
      if (!outReal) outI[g] = scale * ci[v];
    }
  }
}

// ---------------------------------------------------------------------------
// Pointwise kernels
// ---------------------------------------------------------------------------
__global__ void zero_k(float* p, int n) {
  int t = blockIdx.x * blockDim.x + threadIdx.x;
  if (t < n) p[t] = 0.f;
}

// padded + ifftshifted PSF -> real plane
__global__ void psf_pad_shift_k(const float* __restrict__ h, float* __restrict__ outR) {
  int t = blockIdx.x * blockDim.x + threadIdx.x;
  if (t >= HWSZ) return;
  int i = t / FW, j = t % FW;
  int a = (i + FH / 2) % FH;
  int b = (j + FW / 2) % FW;
  float v = 0.f;
  if (a >= P0c && a < P0c + D0c && b >= P1c && b < P1c + D1c)
    v = h[(a - P0c) * D1c + (b - P1c)];
  outR[t] = v;
}

// soft-threshold: u1,u2 = soft(L(sk) + a2/m2, tau)
__global__ void soft_k(const float* __restrict__ sk, const float* __restrict__ a21,
                       const float* __restrict__ a22, const float* __restrict__ mu2,
                       const float* __restrict__ tau, int it,
                       float* __restrict__ u1, float* __restrict__ u2) {
  int t = blockIdx.x * blockDim.x + threadIdx.x;
  if (t >= PLANE) return;
  float m2 = mu2[it], tp = tau[it];
  float inv_m2 = 1.f / m2;
  int hidx = t % HWSZ;
  int i = hidx / FW, j = hidx % FW;
  float v = 0.f, hh = 0.f;
  if (i < FH - 1) v = -(sk[t + FW] - sk[t]) + a21[t] * inv_m2;
  if (j < FW - 1) hh = -(sk[t + 1] - sk[t]) + a22[t] * inv_m2;
  float mag = sqrtf(v * v + hh * hh);
  float magt = fmaxf(mag - tp, 0.f);
  float mm = magt / (magt + tp);
  if (i < FH - 1) u1[t] = v * mm;
  if (j < FW - 1) u2[t] = hh * mm;
}

// vkp = (a1 + m1*Hskp + Cty) / (CtC + m1)
__global__ void vkp_k(const float* __restrict__ a1, const float* __restrict__ Hskp,
                      const float* __restrict__ y, const float* __restrict__ mu1,
                      int it, float* __restrict__ vkp) {
  int t = blockIdx.x * blockDim.x + threadIdx.x;
  if (t >= PLANE) return;
  float m1 = mu1[it];
  int c = t / HWSZ;
  int hidx = t % HWSZ;
  int i = hidx / FW, j = hidx % FW;
  float cty = 0.f, ctc = 0.f;
  if (i >= P0c && i < P0c + D0c && j >= P1c && j < P1c + D1c) {
    cty = y[(c * D0c + (i - P0c)) * D1c + (j - P1c)];
    ctc = 1.f;
  }
  vkp[t] = (a1[t] + m1 * Hskp[t] + cty) / (ctc + m1);
}

// g = vkp - a1/m1  -> real part of complex buffer
__global__ void form_g_k(const float* __restrict__ vkp, const float* __restrict__ a1,
                         const float* __restrict__ mu1, int it, float* __restrict__ gr) {
  int t = blockIdx.x * blockDim.x + threadIdx.x;
  if (t >= PLANE) return;
  gr[t] = vkp[t] - a1[t] / mu1[it];
}

// (outR,outI) = (ar,ai) * (Hr, sign*Hi)   -- complex pointwise, per-channel H reuse
__global__ void cmul_H_k(const float* __restrict__ ar, const float* __restrict__ ai,
                         const float* __restrict__ Hr, const float* __restrict__ Hi,
                         float sign, float* __restrict__ outR, float* __restrict__ outI) {
  int t = blockIdx.x * blockDim.x + threadIdx.x;
  if (t >= PLANE) return;
  int hidx = t % HWSZ;
  float x = ar[t], yv = ai[t];
  float p = Hr[hidx], q = sign * Hi[hidx];
  outR[t] = x * p - yv * q;
  outI[t] = x * q + yv * p;
}

// (outR,outI) = Smult * (ar,ai); Smult = 1/(m1*|H|^2 + m2*LtL + m3), LtL analytic
__global__ void smult_k(const float* __restrict__ ar, const float* __restrict__ ai,
                        const float* __restrict__ Hr, const float* __restrict__ Hi,
                        const float* __restrict__ mu1, const float* __restrict__ mu2,
                        const float* __restrict__ mu3, int it,
                        float* __restrict__ outR, float* __restrict__ outI) {
  int t = blockIdx.x * blockDim.x + threadIdx.x;
  if (t >= PLANE) return;
  float m1 = mu1[it], m2 = mu2[it], m3 = mu3[it];
  int hidx = t % HWSZ;
  int i = hidx / FW, j = hidx % FW;
  float hr = Hr[hidx], hi = Hi[hidx];
  float hth = hr * hr + hi * hi;
  float ltl = 4.f - 2.f * cosf(2.f * PIF * (float)i / (float)FH)
                  - 2.f * cosf(2.f * PIF * (float)j / (float)FW);
  float s = 1.f / (m1 * hth + m2 * ltl + m3);
  outR[t] = s * ar[t];
  outI[t] = s * ai[t];
}

// numer = m3*wkp - a3 + m1*conv + m2*Ltv(u1 - a21/m2, u2 - a22/m2)
__global__ void numer_k(const float* __restrict__ sk, const float* __restrict__ a3,
                        const float* __restrict__ conv, const float* __restrict__ u1,
                        const float* __restrict__ u2, const float* __restrict__ a21,
                        const float* __restrict__ a22, const float* __restrict__ mu1,
                        const float* __restrict__ mu2, const float* __restrict__ mu3,
                        int it, float* __restrict__ numer) {
  int t = blockIdx.x * blockDim.x + threadIdx.x;
  if (t >= PLANE) return;
  float m1 = mu1[it], m2 = mu2[it], m3 = mu3[it];
  float inv_m2 = 1.f / m2, inv_m3 = 1.f / m3;
  int hidx = t % HWSZ;
  int i = hidx / FW, j = hidx % FW;
  // wkp = max(a3/m3 + sk, 0)
  float wkp = fmaxf(a3[t] * inv_m3 + sk[t], 0.f);
  // vertical term
  float va;
  if (i == 0) {
    va = u1[t] - a21[t] * inv_m2;
  } else if (i < FH - 1) {
    float p0 = u1[t] - a21[t] * inv_m2;
    float p1 = u1[t - FW] - a21[t - FW] * inv_m2;
    va = p0 - p1;
  } else {
    va = -(u1[t - FW] - a21[t - FW] * inv_m2);
  }
  // horizontal term
  float vb;
  if (j == 0) {
    vb = u2[t] - a22[t] * inv_m2;
  } else if (j < FW - 1) {
    float q0 = u2[t] - a22[t] * inv_m2;
    float q1 = u2[t - 1] - a22[t - 1] * inv_m2;
    vb = q0 - q1;
  } else {
    vb = -(u2[t - 1] - a22[t - 1] * inv_m2);
  }
  numer[t] = m3 * wkp - a3[t] + m1 * conv[t] + m2 * (va + vb);
}

// dual updates: a1,a21,a22,a3 in place
__global__ void dual_k(const float* __restrict__ skold, const float* __restrict__ skp,
                       const float* __restrict__ vkp, const float* __restrict__ Hskp,
                       const float* __restrict__ u1, const float* __restrict__ u2,
                       float* __restrict__ a1, float* __restrict__ a21,
                       float* __restrict__ a22, float* __restrict__ a3,
                       const float* __restrict__ mu1, const float* __restrict__ mu2,
                       const float* __restrict__ mu3, int it) {
  int t = blockIdx.x * blockDim.x + threadIdx.x;
  if (t >= PLANE) return;
  float m1 = mu1[it], m2 = mu2[it], m3 = mu3[it];
  int hidx = t % HWSZ;
  int i = hidx / FW, j = hidx % FW;
  a1[t] += m1 * (Hskp[t] - vkp[t]);
  if (i < FH - 1) {
    float l1 = -(skp[t + FW] - skp[t]);
    a21[t] += m2 * (l1 - u1[t]);
  }
  if (j < FW - 1) {
    float l2 = -(skp[t + 1] - skp[t]);
    a22[t] += m2 * (l2 - u2[t]);
  }
  float wkp = fmaxf(a3[t] / m3 + skold[t], 0.f);
  a3[t] += m3 * (skp[t] - wkp);
}

// crop + block max
__global__ void crop_max_k(const float* __restrict__ sk, float* __restrict__ partial) {
  __shared__ float red[256];
  int t0 = blockIdx.x * blockDim.x * 4 + threadIdx.x;
  float m = -3.402823e38f;
  for (int u = 0; u < 4; ++u) {
    int o = t0 + u * blockDim.x;
    if (o < CROPSZ) {
      int c = o / (D0c * D1c);
      int rem = o % (D0c * D1c);
      int i = rem / D1c, j = rem % D1c;
      float v = sk[(c * FH + i + P0c) * FW + (j + P1c)];
      m = fmaxf(m, v);
    }
  }
  red[threadIdx.x] = m;
  __syncthreads();
  for (int s = 128; s > 0; s >>= 1) {
    if (threadIdx.x < s) red[threadIdx.x] = fmaxf(red[threadIdx.x], red[threadIdx.x + s]);
    __syncthreads();
  }
  if (threadIdx.x == 0) partial[blockIdx.x] = red[0];
}

__global__ void reduce_max_k(const float* __restrict__ partial, int n, float* __restrict__ mx) {
  __shared__ float red[256];
  float m = -3.402823e38f;
  for (int i = threadIdx.x; i < n; i += 256) m = fmaxf(m, partial[i]);
  red[threadIdx.x] = m;
  __syncthreads();
  for (int s = 128; s > 0; s >>= 1) {
    if (threadIdx.x < s) red[threadIdx.x] = fmaxf(red[threadIdx.x], red[threadIdx.x + s]);
    __syncthreads();
  }
  if (threadIdx.x == 0) mx[0] = red[0];
}

__global__ void normalize_k(const float* __restrict__ sk, const float* __restrict__ mx,
                            float* __restrict__ out) {
  int o = blockIdx.x * blockDim.x + threadIdx.x;
  if (o >= CROPSZ) return;
  int c = o / (D0c * D1c);
  int rem = o % (D0c * D1c);
  int i = rem / D1c, j = rem % D1c;
  float v = sk[(c * FH + i + P0c) * FW + (j + P1c)];
  out[o] = v / mx[0];
}

// ---------------------------------------------------------------------------
// Host orchestration
// ---------------------------------------------------------------------------
extern "C" void kernel_launch(void* const* d_in, const int* in_sizes, int n_in,
                              void* d_out, int out_size, void* d_ws, size_t ws_size,
                              hipStream_t stream) {
  (void)in_sizes; (void)n_in; (void)out_size; (void)ws_size;
  const float* y = (const float*)d_in[0];
  const float* h = (const float*)d_in[1];
  const float* mu1 = (const float*)d_in[2];
  const float* mu2 = (const float*)d_in[3];
  const float* mu3 = (const float*)d_in[4];
  const float* tau = (const float*)d_in[5];
  float* out = (float*)d_out;

  float* w = (float*)d_ws;
  // plane layout: first 6 planes are zero-initialized state
  float* sk   = w + 0L  * PLANE;
  float* a1   = w + 1L  * PLANE;
  float* a3   = w + 2L  * PLANE;
  float* a21  = w + 3L  * PLANE;
  float* a22  = w + 4L  * PLANE;
  float* Hskp = w + 5L  * PLANE;
  float* skB  = w + 6L  * PLANE;
  float* u1   = w + 7L  * PLANE;
  float* u2   = w + 8L  * PLANE;
  float* vkp  = w + 9L  * PLANE;
  float* C1r  = w + 10L * PLANE;
  float* C1i  = w + 11L * PLANE;
  float* C2r  = w + 12L * PLANE;
  float* C2i  = w + 13L * PLANE;
  float* Hr   = w + 14L * PLANE;
  float* Hi   = Hr + HWSZ;
  float* tb   = Hi + HWSZ;
  float* partial = tb + TABLES_SZ;
  float* mxv = partial + 1024;

  const int PW = (PLANE + 255) / 256;
  const int HWG = (HWSZ + 255) / 256;
  const int NB_CROP = (CROPSZ + 1023) / 1024;

  const float* W36F_R = tb + OFF_W36F_R; const float* W36F_I = tb + OFF_W36F_I;
  const float* W36B_R = tb + OFF_W36B_R; const float* W36B_I = tb + OFF_W36B_I;
  const float* W27F_R = tb + OFF_W27F_R; const float* W27F_I = tb + OFF_W27F_I;
  const float* W27B_R = tb + OFF_W27B_R; const float* W27B_I = tb + OFF_W27B_I;
  const float* T36F_R = tb + OFF_T36F_R; const float* T36F_I = tb + OFF_T36F_I;
  const float* T36B_R = tb + OFF_T36B_R; const float* T36B_I = tb + OFF_T36B_I;
  const float* T27F_R = tb + OFF_T27F_R; const float* T27F_I = tb + OFF_T27F_I;
  const float* T27B_R = tb + OFF_T27B_R; const float* T27B_I = tb + OFF_T27B_I;

  const float invN = 1.f / (float)(FH * FW);

  // --- init: tables + zero state ---
  init_tables_k<<<(W36SZ + 255) / 256, 256, 0, stream>>>(tb);
  zero_k<<<(6 * PLANE + 255) / 256, 256, 0, stream>>>(w, 6 * PLANE);

  // --- PSF: H = fft2(ifftshift(pad(h))) -> (Hr,Hi), single plane ---
  psf_pad_shift_k<<<HWG, 256, 0, stream>>>(h, C1r);
  fft1296_k<<<FH, 288, 0, stream>>>(C1r, nullptr, C1r, C1i,
                                    W36F_R, W36F_I, T36F_R, T36F_I, 1, 0, 1.f);
  fft972_k<<<FW, 192, 0, stream>>>(C1r, C1i, Hr, Hi,
                                   W36F_R, W36F_I, W27F_R, W27F_I,
                                   T27F_R, T27F_I, 0, 0, 1.f);

  // --- ADMM iterations ---
  for (int it = 0; it < NITERS; ++it) {
    soft_k<<<PW, 256, 0, stream>>>(sk, a21, a22, mu2, tau, it, u1, u2);
    vkp_k<<<PW, 256, 0, stream>>>(a1, Hskp, y, mu1, it, vkp);

    // conv(vkp - a1/m1, conj(H))  -> C1r (real)
    form_g_k<<<PW, 256, 0, stream>>>(vkp, a1, mu1, it, C1r);
    fft1296_k<<<NC * FH, 288, 0, stream>>>(C1r, nullptr, C1r, C1i,
                                           W36F_R, W36F_I, T36F_R, T36F_I, 1, 0, 1.f);
    fft972_k<<<NC * FW, 192, 0, stream>>>(C1r, C1i, C1r, C1i,
                                          W36F_R, W36F_I, W27F_R, W27F_I,
                                          T27F_R, T27F_I, 0, 0, 1.f);
    cmul_H_k<<<PW, 256, 0, stream>>>(C1r, C1i, Hr, Hi, -1.f, C1r, C1i);
    fft972_k<<<NC * FW, 192, 0, stream>>>(C1r, C1i, C1r, C1i,
                                          W36B_R, W36B_I, W27B_R, W27B_I,
                                          T27B_R, T27B_I, 0, 0, 1.f);
    fft1296_k<<<NC * FH, 288, 0, stream>>>(C1r, C1i, C1r, nullptr,
                                           W36B_R, W36B_I, T36B_R, T36B_I, 0, 1, invN);

    // numer -> skB, then N_hat = fft2(numer) -> C2
    numer_k<<<PW, 256, 0, stream>>>(sk, a3, C1r, u1, u2, a21, a22,
                                    mu1, mu2, mu3, it, skB);
    fft1296_k<<<NC * FH, 288, 0, stream>>>(skB, nullptr, C2r, C2i,
                                           W36F_R, W36F_I, T36F_R, T36F_I, 1, 0, 1.f);
    fft972_k<<<NC * FW, 192, 0, stream>>>(C2r, C2i, C2r, C2i,
                                          W36F_R, W36F_I, W27F_R, W27F_I,
                                          T27F_R, T27F_I, 0, 0, 1.f);

    // Sk_hat = Smult * N_hat -> C1 ; Hskp_hat = H * Sk_hat -> C2
    smult_k<<<PW, 256, 0, stream>>>(C2r, C2i, Hr, Hi, mu1, mu2, mu3, it, C1r, C1i);
    cmul_H_k<<<PW, 256, 0, stream>>>(C1r, C1i, Hr, Hi, 1.f, C2r, C2i);

    // skp = real(ifft2(Sk_hat)) -> skB
    fft972_k<<<NC * FW, 192, 0, stream>>>(C1r, C1i, C1r, C1i,
                                          W36B_R, W36B_I, W27B_R, W27B_I,
                                          T27B_R, T27B_I, 0, 0, 1.f);
    fft1296_k<<<NC * FH, 288, 0, stream>>>(C1r, C1i, skB, nullptr,
                                           W36B_R, W36B_I, T36B_R, T36B_I, 0, 1, invN);

    // Hskp_new = real(ifft2(H * Sk_hat)) -> Hskp
    fft972_k<<<NC * FW, 192, 0, stream>>>(C2r, C2i, C2r, C2i,
                                          W36B_R, W36B_I, W27B_R, W27B_I,
                                          T27B_R, T27B_I, 0, 0, 1.f);
    fft1296_k<<<NC * FH, 288, 0, stream>>>(C2r, C2i, Hskp, nullptr,
                                           W36B_R, W36B_I, T36B_R, T36B_I, 0, 1, invN);

    // dual updates
    dual_k<<<PW, 256, 0, stream>>>(sk, skB, vkp, Hskp, u1, u2,
                                   a1, a21, a22, a3, mu1, mu2, mu3, it);

    // swap sk <-> skB
    float* tmp = sk; sk = skB; skB = tmp;
  }

  // --- crop, global max, normalize ---
  crop_max_k<<<NB_CROP, 256, 0, stream>>>(sk, partial);
  reduce_max_k<<<1, 256, 0, stream>>>(partial, NB_CROP, mxv);
  normalize_k<<<(CROPSZ + 255) / 256, 256, 0, stream>>>(sk, mxv, out);
}